// ModulatedChunks_14431090114849
// MI455X (gfx1250) — compile-verified
//
#include <hip/hip_runtime.h>
#include <hip/hip_bf16.h>

typedef __attribute__((ext_vector_type(2))) float v2f;
typedef __attribute__((ext_vector_type(4))) float v4f;
typedef __attribute__((ext_vector_type(8))) float v8f;

// Problem constants (from setup_inputs)
#define BB 8
#define TT 2048
#define CC 256
#define LL 32
#define DQ 512
#define NW 2017         // T - window + 1
#define SS 2045         // T - cic + 1 distinct chunk-start positions
#define CIC 4           // window/num_chunks

// ---------------------------------------------------------------------------
// Kernel 1: enc{1,2}[b] = query[b] (32x512) @ W (512x256) + bias
// One wave per 16x16 output tile. waves = 8 batches * 2 weights * (2x16 tiles) = 512
// A-frag (16x4 f32): lane L: M = L%16, holds K = kbase + 2*(L>>4) + {0,1}
// B-frag (4x16 f32): lane L: N = L%16, holds K = kbase + 2*(L>>4) + {0,1}
// D (16x16 f32, v8f): lane L, vgpr v: M = 8*(L>>4) + v, N = L%16
// ---------------------------------------------------------------------------
__global__ void enc_gemm_kernel(const float* __restrict__ query,
                                const float* __restrict__ W1, const float* __restrict__ b1,
                                const float* __restrict__ W2, const float* __restrict__ b2,
                                float* __restrict__ enc1, float* __restrict__ enc2) {
    const int lane   = threadIdx.x & 31;
    const int waveId = blockIdx.x * (blockDim.x >> 5) + (threadIdx.x >> 5);
    const int nTile  = waveId & 15;        // 16 tiles along C
    const int mTile  = (waveId >> 4) & 1;  // 2 tiles along L
    const int which  = (waveId >> 5) & 1;  // W1 or W2
    const int b      = waveId >> 6;
    if (b >= BB) return;

    const float* __restrict__ W    = which ? W2 : W1;
    const float* __restrict__ bias = which ? b2 : b1;
    float* __restrict__ enc        = which ? enc2 : enc1;

    const int m = lane & 15;
    const int g = lane >> 4;               // 0 or 1: which K-pair this half-wave holds
    const int n = lane & 15;
    const float* __restrict__ Arow = query + ((size_t)b * LL + mTile * 16 + m) * DQ;
    const int ncol = nTile * 16 + n;

    v8f acc = {0.f, 0.f, 0.f, 0.f, 0.f, 0.f, 0.f, 0.f};
    for (int k = 0; k < DQ; k += 4) {
        v2f a;
        a.x = Arow[k + 2 * g + 0];
        a.y = Arow[k + 2 * g + 1];
        v2f bm;
        bm.x = W[(size_t)(k + 2 * g + 0) * CC + ncol];
        bm.y = W[(size_t)(k + 2 * g + 1) * CC + ncol];
        acc = __builtin_amdgcn_wmma_f32_16x16x4_f32(false, a, false, bm,
                                                    (short)0, acc, false, false);
    }
#pragma unroll
    for (int v = 0; v < 8; ++v) {
        int row = mTile * 16 + 8 * g + v;
        enc[((size_t)b * LL + row) * CC + ncol] = acc[v] + bias[ncol];
    }
}

// ---------------------------------------------------------------------------
// Kernel 2: sim[b] = vis[b] (2048x256) @ enc1[b]^T (256x32)
// waves = 8 * 128 Mtiles * 2 Ntiles = 2048. B[k=c, n=l] = enc1[l, c] (rows of
// enc1 are contiguous in k => float2 loads for both fragments).
// ---------------------------------------------------------------------------
__global__ void sim_gemm_kernel(const float* __restrict__ vis,
                                const float* __restrict__ enc1,
                                float* __restrict__ sim) {
    const int lane   = threadIdx.x & 31;
    const int waveId = blockIdx.x * (blockDim.x >> 5) + (threadIdx.x >> 5);
    const int nTile  = waveId & 1;           // 2 tiles along L
    const int mTile  = (waveId >> 1) & 127;  // 128 tiles along T
    const int b      = waveId >> 8;
    if (b >= BB) return;

    const int m = lane & 15;
    const int g = lane >> 4;
    const int n = lane & 15;
    const float* __restrict__ Arow = vis  + ((size_t)b * TT + mTile * 16 + m) * CC;
    const float* __restrict__ Brow = enc1 + ((size_t)b * LL + nTile * 16 + n) * CC;

    v8f acc = {0.f, 0.f, 0.f, 0.f, 0.f, 0.f, 0.f, 0.f};
    for (int k = 0; k < CC; k += 4) {
        v2f a;
        a.x = Arow[k + 2 * g + 0];
        a.y = Arow[k + 2 * g + 1];
        v2f bm;
        bm.x = Brow[k + 2 * g + 0];
        bm.y = Brow[k + 2 * g + 1];
        acc = __builtin_amdgcn_wmma_f32_16x16x4_f32(false, a, false, bm,
                                                    (short)0, acc, false, false);
    }
#pragma unroll
    for (int v = 0; v < 8; ++v) {
        int t = mTile * 16 + 8 * g + v;
        sim[((size_t)b * TT + t) * LL + nTile * 16 + n] = acc[v];
    }
}

// ---------------------------------------------------------------------------
// Kernel 3: argmax over L=32 (first max, matching jnp.argmax)
// ---------------------------------------------------------------------------
__global__ void argmax_kernel(const float* __restrict__ sim, int* __restrict__ labels) {
    int idx = blockIdx.x * blockDim.x + threadIdx.x;  // b*T + t
    if (idx >= BB * TT) return;
    const float* __restrict__ row = sim + (size_t)idx * LL;
    float best = row[0];
    int bi = 0;
#pragma unroll
    for (int l = 1; l < LL; ++l) {
        float v = row[l];
        if (v > best) { best = v; bi = l; }
    }
    labels[idx] = bi;
}

// ---------------------------------------------------------------------------
// Kernel 4: fused[b,s,:] = enc2[b, majority(labels[b,s..s+3]), :]
//                          * mean(vis[b, s..s+3, :]),  s in [0, 2044]
// 64 threads (v4f) per row, 4 rows per 256-thread block.
// Majority with tie -> smallest label == argmax over one-hot counts.
// ---------------------------------------------------------------------------
__global__ void fused_rows_kernel(const float* __restrict__ vis,
                                  const float* __restrict__ enc2,
                                  const int* __restrict__ labels,
                                  float* __restrict__ fused) {
    const int row  = blockIdx.x * (blockDim.x >> 6) + (threadIdx.x >> 6);
    const int lane = threadIdx.x & 63;
    if (row >= BB * SS) return;
    const int b = row / SS;
    const int s = row - b * SS;

    const int* __restrict__ lab = labels + b * TT + s;
    int ls0 = lab[0], ls1 = lab[1], ls2 = lab[2], ls3 = lab[3];
    int ls[4] = {ls0, ls1, ls2, ls3};
    int bestLab = ls0, bestCnt = 0;
#pragma unroll
    for (int i = 0; i < 4; ++i) {
        int c = 0;
#pragma unroll
        for (int j = 0; j < 4; ++j) c += (ls[j] == ls[i]) ? 1 : 0;
        if (c > bestCnt || (c == bestCnt && ls[i] < bestLab)) { bestCnt = c; bestLab = ls[i]; }
    }

    const v4f* __restrict__ e  = (const v4f*)(enc2 + ((size_t)b * LL + bestLab) * CC);
    const v4f* __restrict__ v0 = (const v4f*)(vis + ((size_t)b * TT + s) * CC);
    v4f a0 = v0[lane];
    v4f a1 = v0[lane + 1 * (CC / 4)];
    v4f a2 = v0[lane + 2 * (CC / 4)];
    v4f a3 = v0[lane + 3 * (CC / 4)];
    v4f ee = e[lane];
    v4f r = ee * 0.25f * (a0 + a1 + a2 + a3);
    ((v4f*)(fused + (size_t)row * CC))[lane] = r;
}

// ---------------------------------------------------------------------------
// Kernel 5: out[b,w,j,:] = fused[b, w+4j, :]  (streaming broadcast)
// fused (16.7 MB) lives in L2; output stores are non-temporal.
// ---------------------------------------------------------------------------
__global__ void expand_kernel(const float* __restrict__ fused, float* __restrict__ out) {
    unsigned idx = blockIdx.x * blockDim.x + threadIdx.x;  // one v4f each
    const unsigned total = (unsigned)BB * NW * 8 * (CC / 4);
    if (idx >= total) return;
    unsigned c4   = idx & 63;       // CC/4 = 64
    unsigned rest = idx >> 6;       // (b*NW + w)*8 + j
    unsigned j    = rest & 7;
    unsigned bw   = rest >> 3;      // b*NW + w
    unsigned w    = bw % NW;
    unsigned b    = bw / NW;
    unsigned s    = w + 4u * j;
    v4f val = ((const v4f*)(fused + ((size_t)b * SS + s) * CC))[c4];
    __builtin_nontemporal_store(val, ((v4f*)out) + idx);
}

// ---------------------------------------------------------------------------
extern "C" void kernel_launch(void* const* d_in, const int* in_sizes, int n_in,
                              void* d_out, int out_size, void* d_ws, size_t ws_size,
                              hipStream_t stream) {
    (void)in_sizes; (void)n_in; (void)out_size; (void)ws_size;
    const float* vis   = (const float*)d_in[0];
    const float* query = (const float*)d_in[1];
    const float* W1    = (const float*)d_in[2];
    const float* b1    = (const float*)d_in[3];
    const float* W2    = (const float*)d_in[4];
    const float* b2    = (const float*)d_in[5];
    // d_in[6] = window_size (32), d_in[7] = num_chunks (8): baked as constants.

    float* ws    = (float*)d_ws;
    float* enc1  = ws;                     // 8*32*256            = 65536 f
    float* enc2  = ws + 65536;             // 65536 f
    float* sim   = ws + 131072;            // 8*2048*32           = 524288 f
    int*   labels= (int*)(ws + 655360);    // 8*2048              = 16384 i32
    float* fused = ws + 671744;            // 8*2045*256          = 4188160 f
    float* out   = (float*)d_out;

    // 1) enc GEMMs: 512 waves, 4 waves/block
    enc_gemm_kernel<<<128, 128, 0, stream>>>(query, W1, b1, W2, b2, enc1, enc2);
    // 2) sim GEMM: 2048 waves, 8 waves/block
    sim_gemm_kernel<<<256, 256, 0, stream>>>(vis, enc1, sim);
    // 3) argmax: 16384 threads
    argmax_kernel<<<64, 256, 0, stream>>>(sim, labels);
    // 4) fused rows: 8*2045 rows, 4 rows/block
    {
        int rows = BB * SS;
        int blocks = (rows + 3) / 4;
        fused_rows_kernel<<<blocks, 256, 0, stream>>>(vis, enc2, labels, fused);
    }
    // 5) expand: 8*2017*8*64 v4f stores
    {
        unsigned total = (unsigned)BB * NW * 8 * (CC / 4);
        unsigned blocks = (total + 255u) / 256u;
        expand_kernel<<<blocks, 256, 0, stream>>>(fused, out);
    }
}